// GaussianScorer_58961311040187
// MI455X (gfx1250) — compile-verified
//
#include <hip/hip_runtime.h>
#include <hip/hip_bf16.h>
#include <math.h>

typedef __attribute__((ext_vector_type(2))) float v2f;
typedef __attribute__((ext_vector_type(4))) float v4f;
typedef __attribute__((ext_vector_type(8))) float v8f;

#define E_DIM   512
#define L_DIM   128
#define KCHUNK  256
#define PRSTRIDE 144                      // pair-row stride in v2f (288 dwords = 32 mod 64 banks)

// workspace layout (float offsets)
// Btp: pair-interleaved, float2 at [pp*128 + l] = { means[l][2pp]*iv, means[l][2pp+1]*iv }
#define WS_BTP   0                        // 256 pair-rows * 128 * 2 = 65536 floats
#define WS_IVAR  (E_DIM * L_DIM)          // inv_var[512]
#define WS_M2    (WS_IVAR + E_DIM)        // m2[128]
#define WS_CONST (WS_M2 + L_DIM)          // scalar: factor - 0.5*sum(log var)

// ---------------------------------------------------------------------------
// prep: inv_var, Btp (transposed, scaled, K-pair interleaved), m2, const term
// ---------------------------------------------------------------------------
__global__ __launch_bounds__(256) void GaussianScorer_prep(
    const float* __restrict__ means, const float* __restrict__ var,
    float* __restrict__ ws) {
  __shared__ float sIv[E_DIM];
  __shared__ float sRed[256];
  const int tid = threadIdx.x;

  float slog = 0.0f;
  for (int e = tid; e < E_DIM; e += 256) {
    float v  = var[e];
    float iv = 1.0f / v;
    sIv[e] = iv;
    ws[WS_IVAR + e] = iv;
    slog += logf(v);
  }
  sRed[tid] = slog;
  __syncthreads();
  for (int s = 128; s > 0; s >>= 1) {
    if (tid < s) sRed[tid] += sRed[tid + s];
    __syncthreads();
  }
  if (tid == 0) {
    const float log2pi = 1.8378770664093453f;
    const float factor = -0.5f * (float)E_DIM * log2pi;
    ws[WS_CONST] = factor - 0.5f * sRed[0];
  }

  // Btp[(pp*128 + l)*2 + q] = means[l][2pp+q] * iv[2pp+q]   (coalesced store)
  for (int idx = tid; idx < E_DIM * L_DIM; idx += 256) {
    int pp  = idx >> 8;          // K-pair index 0..255
    int rem = idx & 255;
    int l   = rem >> 1;
    int q   = rem & 1;
    int e   = 2 * pp + q;
    ws[WS_BTP + idx] = means[l * E_DIM + e] * sIv[e];
  }

  // m2[l] = sum_e means[l][e]^2 * inv_var[e]
  if (tid < L_DIM) {
    float s = 0.0f;
    for (int e = 0; e < E_DIM; ++e) {
      float m = means[tid * E_DIM + e];
      s += m * m * sIv[e];
    }
    ws[WS_M2 + tid] = s;
  }
}

// ---------------------------------------------------------------------------
// main: 4 waves/block, each wave -> 16 rows x 128 cols via 8 WMMA f32 tiles
// ---------------------------------------------------------------------------
__global__ __launch_bounds__(128) void GaussianScorer_gemm(
    const float* __restrict__ emb, const float* __restrict__ ws,
    float* __restrict__ out) {
  __shared__ v2f   sBp[(KCHUNK / 2) * PRSTRIDE];   // 128 pair-rows, padded: 144 KB
  __shared__ float sIv[KCHUNK];                    // inv_var chunk
  __shared__ float sE2[64];                        // per-row e2, 4 waves * 16 rows

  const int tid   = threadIdx.x;
  const int wave  = tid >> 5;
  const int lane  = tid & 31;
  const int col   = lane & 15;           // N index within 16-wide tile
  const int khalf = lane >> 4;           // 0: K pair 0, 1: K pair 1 of each step
  const int kgrp  = khalf * 2;
  const int rowStart = blockIdx.x * 64 + wave * 16;
  const int myRow    = rowStart + col;   // A-matrix row this lane feeds

  const float* aRow = emb + (size_t)myRow * E_DIM;

  v8f acc[8] = {};
  float e2p = 0.0f;

  for (int chunk = 0; chunk < 2; ++chunk) {
    // ---- stage Btp chunk (b128 coalesced -> padded LDS rows) + inv_var chunk
    const v4f* __restrict__ srcB =
        (const v4f*)(ws + WS_BTP + chunk * (KCHUNK / 2) * (L_DIM * 2));
    for (int j = tid; j < (KCHUNK / 2) * (L_DIM / 2); j += 128) {
      int pr = j >> 6;                   // pair-row 0..127
      int q  = j & 63;                   // quad within row
      *((v4f*)(sBp + pr * PRSTRIDE) + q) = srcB[pr * (L_DIM / 2) + q];
    }
    if (tid < KCHUNK / 4)
      ((v4f*)sIv)[tid] = ((const v4f*)(ws + WS_IVAR + chunk * KCHUNK))[tid];
    __syncthreads();

    const float* aC = aRow + chunk * KCHUNK + kgrp;
#pragma unroll 4
    for (int i = 0; i < KCHUNK / 4; ++i) {
      const int kl = i * 4 + kgrp;
      // A operand: lane (0-15) holds A[m][k],A[m][k+1]; lane+16 holds k+2,k+3
      v2f a  = *(const v2f*)(aC + i * 4);
      v2f iv = *(const v2f*)(sIv + kl);          // one ds_load_b64
      e2p += a.x * a.x * iv.x + a.y * a.y * iv.y;

      // B operand for this lane-half: pair-row = i*2 + khalf, single b64/tile
      const v2f* bRow = sBp + (i * 2 + khalf) * PRSTRIDE + col;
#pragma unroll
      for (int t = 0; t < 8; ++t) {
        v2f b = bRow[t * 16];                    // {B[k][n], B[k+1][n]}
        acc[t] = __builtin_amdgcn_wmma_f32_16x16x4_f32(
            false, a, false, b, (short)0, acc[t], false, false);
      }
    }
    __syncthreads();
  }

  // ---- combine e2 halves (K split across lane groups) and share via LDS
  float e2t = e2p + __shfl_xor(e2p, 16, 32);
  if (lane < 16) sE2[wave * 16 + col] = e2t;
  __syncthreads();

  const float cTerm = ws[WS_CONST];
#pragma unroll
  for (int t = 0; t < 8; ++t) {
    float m2c = ws[WS_M2 + t * 16 + col];
#pragma unroll
    for (int i = 0; i < 8; ++i) {
      // C/D layout: VGPR i -> row khalf*8 + i, col = lane&15
      int rl = khalf * 8 + i;
      float val = cTerm + acc[t][i] - 0.5f * (sE2[wave * 16 + rl] + m2c);
      out[(size_t)(rowStart + rl) * L_DIM + t * 16 + col] = val;
    }
  }
}

// ---------------------------------------------------------------------------
extern "C" void kernel_launch(void* const* d_in, const int* in_sizes, int n_in,
                              void* d_out, int out_size, void* d_ws, size_t ws_size,
                              hipStream_t stream) {
  const float* emb   = (const float*)d_in[0];  // [8,2048,512]
  const float* means = (const float*)d_in[1];  // [128,512]
  const float* var   = (const float*)d_in[2];  // [512]
  float* out = (float*)d_out;                  // [8,2048,128]
  float* ws  = (float*)d_ws;                   // needs ~259 KB

  GaussianScorer_prep<<<1, 256, 0, stream>>>(means, var, ws);
  // 16384 rows / 64 rows per block = 256 blocks
  GaussianScorer_gemm<<<256, 128, 0, stream>>>(emb, ws, out);
}